// SparseBBLayer_30837865185920
// MI455X (gfx1250) — compile-verified
//
#include <hip/hip_runtime.h>

typedef float v2f __attribute__((ext_vector_type(2)));
typedef float v8f __attribute__((ext_vector_type(8)));
typedef int   v4i __attribute__((ext_vector_type(4)));

#define GLOBAL_AS __attribute__((address_space(1)))
#define LDS_AS    __attribute__((address_space(3)))

#define D_INP 4096
#define D_OUT 4096
#define BLK_N 128          // x-rows per workgroup
#define BLK_M 64           // out-cols per workgroup
#define KC    32           // K chunk staged in LDS
#define LDSS  36           // padded row stride (floats): conflict-free, 16B-aligned
#define NCHUNK (D_INP / KC)

#if __has_builtin(__builtin_amdgcn_global_load_async_to_lds_b128)
#define HAVE_ASYNC_LDS 1
#else
#define HAVE_ASYNC_LDS 0
#endif

__device__ __forceinline__ void wait_async_zero() {
#if HAVE_ASYNC_LDS
#if __has_builtin(__builtin_amdgcn_s_wait_asynccnt)
    __builtin_amdgcn_s_wait_asynccnt(0);
#else
    asm volatile("s_wait_asynccnt 0x0" ::: "memory");
#endif
#endif
}

// out = x @ (ws*mask + wb)^T + bias   (N=8192, M=4096, K=4096, fp32)
// Pure HBM-bandwidth-bound (~470MB @ 23.3 TB/s): weights stay L2-resident and
// are fused (ws*mask+wb) at LDS-staging time; x tile is DMA'd with
// global_load_async_to_lds_b128 into double-buffered LDS; output is streamed
// with non-temporal stores so it cannot evict the L2-resident weight streams.
__global__ __launch_bounds__(256)
void sparse_bb_gemm(const float* __restrict__ x,
                    const float* __restrict__ ws,
                    const float* __restrict__ mask,
                    const float* __restrict__ wb,
                    const float* __restrict__ bias,
                    float* __restrict__ out)
{
    __shared__ float Alds[2][BLK_N * LDSS];   // x tiles      128 x 32, ping-pong
    __shared__ float Blds[2][BLK_M * LDSS];   // weight tiles  64 x 32, ping-pong

    const int tid  = threadIdx.x;
    const int lane = tid & 31;
    const int wave = tid >> 5;                 // 0..7
    const int wN   = wave & 3;                 // 4 waves along N
    const int wM   = wave >> 2;                // 2 waves along M
    const int rowBase = blockIdx.y * BLK_N;
    const int colBase = blockIdx.x * BLK_M;

    const int l    = lane & 15;                // fragment row within 16
    const int koff = (lane >> 4) << 1;         // K sub-offset: 0 / 2 per lane half

    v8f acc[2][2];
#pragma unroll
    for (int i = 0; i < 2; ++i)
#pragma unroll
        for (int j = 0; j < 2; ++j)
            acc[i][j] = (v8f){0.f,0.f,0.f,0.f,0.f,0.f,0.f,0.f};

    // ---- x-tile staging: 128 rows x 32 K = 1024 float4, 4 per thread ----
    auto stageA = [&](int k0, int buf) {
#pragma unroll
        for (int i = 0; i < 4; ++i) {
            const int idx = tid + i * 256;           // 0..1023
            const int r   = idx >> 3;                // 0..127
            const int s   = idx & 7;                 // 0..7
            const size_t g = (size_t)(rowBase + r) * D_INP + k0 + s * 4;
            float* dst = &Alds[buf][r * LDSS + s * 4];
#if HAVE_ASYNC_LDS
            __builtin_amdgcn_global_load_async_to_lds_b128(
                (GLOBAL_AS v4i*)&x[g], (LDS_AS v4i*)dst, 0, 0);
#else
            *(float4*)dst = *(const float4*)&x[g];
#endif
        }
    };

    // ---- weight-tile staging (fused ws*mask + wb): 64 rows x 32 K ----
    float4 wsr[2], mr[2], wbr[2];
    auto loadB = [&](int k0) {                       // issue global loads -> regs
#pragma unroll
        for (int i = 0; i < 2; ++i) {
            const int idx = tid + i * 256;           // 0..511
            const int r   = idx >> 3;                // 0..63
            const int s   = idx & 7;
            const size_t g = (size_t)(colBase + r) * D_INP + k0 + s * 4;
            wsr[i] = *(const float4*)&ws[g];
            mr[i]  = *(const float4*)&mask[g];
            wbr[i] = *(const float4*)&wb[g];
        }
    };
    auto commitB = [&](int buf) {                    // fuse + store to LDS
#pragma unroll
        for (int i = 0; i < 2; ++i) {
            const int idx = tid + i * 256;
            const int r   = idx >> 3;
            const int s   = idx & 7;
            float4 o;
            o.x = fmaf(wsr[i].x, mr[i].x, wbr[i].x);
            o.y = fmaf(wsr[i].y, mr[i].y, wbr[i].y);
            o.z = fmaf(wsr[i].z, mr[i].z, wbr[i].z);
            o.w = fmaf(wsr[i].w, mr[i].w, wbr[i].w);
            *(float4*)&Blds[buf][r * LDSS + s * 4] = o;
        }
    };

    // ---- prologue: fill buffer 0 ----
    loadB(0);
    commitB(0);
    stageA(0, 0);
    wait_async_zero();
    __syncthreads();

    // ---- main pipeline: prefetch chunk c+1 while computing chunk c ----
    for (int c = 0; c < NCHUNK; ++c) {
        const int  buf  = c & 1;
        const int  nbuf = buf ^ 1;
        const bool more = (c + 1 < NCHUNK);
        if (more) {
            loadB((c + 1) * KC);                     // weight loads in flight
            stageA((c + 1) * KC, nbuf);              // async DMA into other buffer
        }

        // 8 K-steps x (2x2) tiles of v_wmma_f32_16x16x4_f32 = 32 WMMAs
#pragma unroll
        for (int kk = 0; kk < KC; kk += 4) {
            v2f aF[2], bF[2];
#pragma unroll
            for (int sn = 0; sn < 2; ++sn) {
                const int r = wN * 32 + sn * 16 + l;
                aF[sn] = *(const v2f*)&Alds[buf][r * LDSS + kk + koff];
            }
#pragma unroll
            for (int sm = 0; sm < 2; ++sm) {
                const int r = wM * 32 + sm * 16 + l;
                bF[sm] = *(const v2f*)&Blds[buf][r * LDSS + kk + koff];
            }
#pragma unroll
            for (int sn = 0; sn < 2; ++sn)
#pragma unroll
                for (int sm = 0; sm < 2; ++sm)
                    acc[sn][sm] = __builtin_amdgcn_wmma_f32_16x16x4_f32(
                        false, aF[sn], false, bF[sm],
                        (short)0, acc[sn][sm], false, false);
        }

        if (more) commitB(nbuf);                     // fuse+store under WMMA shadow
        wait_async_zero();
        __syncthreads();
    }

    // ---- epilogue: D[r + 8*(lane>=16)][lane&15] per ISA C layout, + bias ----
    // Non-temporal: out is write-once, keep it out of the weight-resident L2.
    const int hi = (lane >> 4) * 8;
#pragma unroll
    for (int sn = 0; sn < 2; ++sn) {
#pragma unroll
        for (int sm = 0; sm < 2; ++sm) {
            const int j  = colBase + wM * 32 + sm * 16 + l;
            const float bj = bias[j];
            const int ibase = rowBase + wN * 32 + sn * 16 + hi;
#pragma unroll
            for (int r = 0; r < 8; ++r)
                __builtin_nontemporal_store(acc[sn][sm][r] + bj,
                                            &out[(size_t)(ibase + r) * D_OUT + j]);
        }
    }
}

extern "C" void kernel_launch(void* const* d_in, const int* in_sizes, int n_in,
                              void* d_out, int out_size, void* d_ws, size_t ws_size,
                              hipStream_t stream) {
    const float* x    = (const float*)d_in[0];
    const float* wsp  = (const float*)d_in[1];
    const float* msk  = (const float*)d_in[2];
    const float* wbb  = (const float*)d_in[3];
    const float* bias = (const float*)d_in[4];
    float* out = (float*)d_out;

    const int Nrows = in_sizes[0] / D_INP;   // 8192
    dim3 grid(D_OUT / BLK_M, Nrows / BLK_N); // 64 x 64
    sparse_bb_gemm<<<grid, dim3(256), 0, stream>>>(x, wsp, msk, wbb, bias, out);
}